// GraphAttention_89318139887750
// MI455X (gfx1250) — compile-verified
//
#include <hip/hip_runtime.h>
#include <stdint.h>

// GAT forward, B=4, N=4096, F=64, U=32, H=4 — gfx1250 (MI455X).
// f32 WMMA for both GEMMs; one-pass streaming softmax (lrelu is monotone, so
// rowmax = lrelu(s_n + max_m t_m)); async global->LDS double-buffered staging.
//
// feats workspace layout is pair-interleaved: element (m,u) lives at
//   base[y] + (m>>1)*64 + (u<<1) + (m&1)
// so a 16-byte global chunk == 16-byte LDS chunk (async B128 friendly) and a
// WMMA B-operand K-pair (kk,kk+1) at column u is one aligned ds_load_b64.

typedef __attribute__((ext_vector_type(2))) float v2f;
typedef __attribute__((ext_vector_type(8))) float v8f;

#define B_ 4
#define N_ 4096
#define F_ 64
#define U_ 32
#define H_ 4
#define ALPHA 0.2f
#define BM 128                  // m-chunk per LDS buffer
#define NCHUNK (N_ / BM)

__device__ __forceinline__ float lrelu(float e) {
    return fmaxf(e, ALPHA * e); // both slopes positive
}

__device__ __forceinline__ uint32_t lds_off32(const void* p) {
    // low 32 bits of a generic shared-aperture address == wave LDS byte offset
    return (uint32_t)(uintptr_t)p;
}

__device__ __forceinline__ void async_copy_b128(uint32_t lds_byte_off,
                                                const float* gbase,
                                                uint32_t gbyte_off) {
    // GLOBAL_LOAD_ASYNC_TO_LDS_B128, GVS mode: mem = SADDR + VADDR; tracked by ASYNCcnt
    asm volatile("global_load_async_to_lds_b128 %0, %1, %2"
                 :: "v"(lds_byte_off), "v"(gbyte_off), "s"(gbase)
                 : "memory");
}

// ---------------------------------------------------------------------------
// Kernel A: feats[b,h] = x[b] @ kernels[h]  ([4096x64]x[64x32]), WMMA f32 16x16x4.
// grid = (N/128, B*H), block = 256; wave -> one 16x32 tile. Weights staged
// swizzled in LDS so B operands are ds_load_b64. Output written pair-interleaved.
// ---------------------------------------------------------------------------
__global__ __launch_bounds__(256) void gat_proj(const float* __restrict__ x,
                                                const float* __restrict__ kernels,
                                                float* __restrict__ feats) {
    __shared__ __align__(16) float klds[F_ * U_];   // [F/2][U][2] interleaved

    const int y    = blockIdx.y;            // b*H + h
    const int b    = y / H_;
    const int h    = y % H_;
    const int tid  = threadIdx.x;
    const int lane = tid & 31;
    const int wave = tid >> 5;
    const int lr   = lane & 15;
    const int hi   = lane >> 4;
    const int rowTile = blockIdx.x * 128 + wave * 16;

    // stage kernels[h] swizzled: (k,u) -> klds[(k>>1)*64 + (u<<1) + (k&1)]
    const float* kh = kernels + (size_t)h * F_ * U_;
    #pragma unroll
    for (int j = 0; j < (F_ * U_) / 256; ++j) {
        const int g = tid + j * 256;        // coalesced global index = k*32+u
        const int k = g >> 5, u = g & 31;
        klds[(k >> 1) * 64 + (u << 1) + (k & 1)] = kh[g];
    }
    __syncthreads();

    const float* xrow = x + ((size_t)b * N_ + (rowTile + lr)) * F_;

    v8f acc0 = {}; v8f acc1 = {};
    #pragma unroll
    for (int k0 = 0; k0 < F_; k0 += 4) {
        const int kk = k0 + 2 * hi;                         // A VGPR j: K = j + 2*hi
        const v2f a  = *(const v2f*)(xrow + kk);            // global b64
        const v2f b0 = *(const v2f*)(klds + (kk >> 1) * 64 + (lr << 1));        // ds b64
        const v2f b1 = *(const v2f*)(klds + (kk >> 1) * 64 + (lr << 1) + 32);   // ds b64
        acc0 = __builtin_amdgcn_wmma_f32_16x16x4_f32(false, a, false, b0, (short)0, acc0, false, false);
        acc1 = __builtin_amdgcn_wmma_f32_16x16x4_f32(false, a, false, b1, (short)0, acc1, false, false);
    }

    float* fout = feats + (size_t)y * N_ * U_;
    #pragma unroll
    for (int i = 0; i < 8; ++i) {           // C/D: row = i + 8*hi, col = lr
        const int n = rowTile + i + 8 * hi;
        const int pr = (n >> 1) * 64 + (n & 1);
        fout[pr + (lr << 1)]        = acc0[i];
        fout[pr + ((lr + 16) << 1)] = acc1[i];
    }
}

// ---------------------------------------------------------------------------
// Kernel C: s/t logits from (pair-interleaved) feats. One thread per node.
// ---------------------------------------------------------------------------
__global__ __launch_bounds__(256) void gat_logits(const float* __restrict__ feats,
                                                  const float* __restrict__ att_self,
                                                  const float* __restrict__ att_neigh,
                                                  float* __restrict__ s,
                                                  float* __restrict__ t) {
    const int idx = blockIdx.x * 256 + threadIdx.x;   // y*N + n
    const int y   = idx / N_;
    const int h   = y % H_;
    const int n   = idx & (N_ - 1);
    const float* f = feats + (size_t)y * N_ * U_ + (n >> 1) * 64 + (n & 1);
    float ss = 0.f, tt = 0.f;
    #pragma unroll
    for (int u = 0; u < U_; ++u) {
        const float fv = f[u << 1];
        ss += fv * att_self [h * U_ + u];
        tt += fv * att_neigh[h * U_ + u];
    }
    s[idx] = ss;
    t[idx] = tt;
}

// ---------------------------------------------------------------------------
// Kernel D: tmax[y] = max_n t[y,n]
// ---------------------------------------------------------------------------
__global__ __launch_bounds__(256) void gat_tmax(const float* __restrict__ t,
                                                float* __restrict__ tmax) {
    __shared__ float red[256];
    const float* ty = t + (size_t)blockIdx.x * N_;
    float m = -3.4e38f;
    for (int i = threadIdx.x; i < N_; i += 256) m = fmaxf(m, ty[i]);
    red[threadIdx.x] = m;
    __syncthreads();
    for (int off = 128; off > 0; off >>= 1) {
        if (threadIdx.x < off)
            red[threadIdx.x] = fmaxf(red[threadIdx.x], red[threadIdx.x + off]);
        __syncthreads();
    }
    if (threadIdx.x == 0) tmax[blockIdx.x] = red[0];
}

// ---------------------------------------------------------------------------
// Kernel B: streaming softmax attention, async double-buffered LDS staging.
// grid = (N/128, B*H), block = 256 (8 waves); wave owns 16 rows.
// ---------------------------------------------------------------------------
__global__ __launch_bounds__(256) void gat_attn(const float* __restrict__ feats,
                                                const float* __restrict__ s,
                                                const float* __restrict__ t,
                                                const float* __restrict__ tmax,
                                                const float* __restrict__ biases,
                                                float* __restrict__ out) {
    __shared__ __align__(16) float fl[2][BM * U_];  // 2 x 16 KB chunk buffers
    __shared__ float zsh[8 * 16];

    const int y    = blockIdx.y;
    const int b    = y / H_;
    const int h    = y % H_;
    const int tid  = threadIdx.x;
    const int lane = tid & 31;
    const int wave = tid >> 5;
    const int lr   = lane & 15;
    const int hi   = lane >> 4;
    const int rowTile = blockIdx.x * 128 + wave * 16;

    const float sRow = s[(size_t)y * N_ + rowTile + lr];
    const float eM   = lrelu(sRow + tmax[y]);       // exact row max

    const float* fy = feats + (size_t)y * N_ * U_;  // pair-interleaved chunk stream
    const float* ty = t + (size_t)y * N_;

    const uint32_t flo[2] = { lds_off32(&fl[0][0]), lds_off32(&fl[1][0]) };
    const uint32_t CHUNK_BYTES = BM * U_ * 4;       // 16 KB

    // prologue: async-stage chunk 0 into buffer 0 (4 x B128 per wave)
    #pragma unroll
    for (int i = 0; i < 4; ++i) {
        const uint32_t o = (uint32_t)(tid + i * 256) * 16;
        async_copy_b128(flo[0] + o, fy, o);
    }

    v8f acc0 = {}, acc1 = {};
    float z = 0.f;
    int buf = 0;

    for (int c = 0; c < NCHUNK; ++c) {
        const int m0 = c * BM;
        if (c + 1 < NCHUNK) {
            // prefetch next chunk into the other buffer, then wait for current
            const uint32_t gb = (uint32_t)(c + 1) * CHUNK_BYTES;
            #pragma unroll
            for (int i = 0; i < 4; ++i) {
                const uint32_t o = (uint32_t)(tid + i * 256) * 16;
                async_copy_b128(flo[buf ^ 1] + o, fy, gb + o);
            }
            asm volatile("s_wait_asynccnt 0x4" ::: "memory"); // current chunk done (in-order)
        } else {
            asm volatile("s_wait_asynccnt 0x0" ::: "memory");
        }
        __syncthreads();                             // all waves' DMA visible

        const float* fb = &fl[buf][0];
        #pragma unroll 8
        for (int mk = 0; mk < BM; mk += 4) {
            const int kk = mk + 2 * hi;              // this lane's K pair
            const v2f tp = *(const v2f*)(ty + m0 + kk);          // global b64, L2-hot
            const float p0 = __expf(lrelu(sRow + tp.x) - eM);
            const float p1 = __expf(lrelu(sRow + tp.y) - eM);
            z += p0 + p1;
            v2f a; a.x = p0; a.y = p1;
            const v2f b0 = *(const v2f*)(fb + (kk >> 1) * 64 + (lr << 1));       // ds b64
            const v2f b1 = *(const v2f*)(fb + (kk >> 1) * 64 + (lr << 1) + 32);  // ds b64
            acc0 = __builtin_amdgcn_wmma_f32_16x16x4_f32(false, a, false, b0, (short)0, acc0, false, false);
            acc1 = __builtin_amdgcn_wmma_f32_16x16x4_f32(false, a, false, b1, (short)0, acc1, false, false);
        }
        __syncthreads();                             // reads done before buffer reuse
        buf ^= 1;
    }

    // combine the two lanes holding each row, publish Z per row
    z += __shfl_xor(z, 16, 32);
    if (hi == 0) zsh[wave * 16 + lr] = z;
    __syncthreads();

    #pragma unroll
    for (int i = 0; i < 8; ++i) {
        const int lrow = i + 8 * hi;                 // local row of C/D element i
        const float zr = zsh[wave * 16 + lrow];
        const int n = rowTile + lrow;
        float* o = out + ((size_t)b * N_ + n) * (H_ * U_) + h * U_;
        const float v0 = acc0[i] / zr + biases[h * U_ + lr];
        const float v1 = acc1[i] / zr + biases[h * U_ + 16 + lr];
        o[lr]      = fmaxf(v0, 0.f);
        o[16 + lr] = fmaxf(v1, 0.f);
    }
}

// ---------------------------------------------------------------------------
extern "C" void kernel_launch(void* const* d_in, const int* in_sizes, int n_in,
                              void* d_out, int out_size, void* d_ws, size_t ws_size,
                              hipStream_t stream) {
    const float* x         = (const float*)d_in[0];   // [B,N,F]
    const float* kernels   = (const float*)d_in[1];   // [H,F,U]
    const float* att_self  = (const float*)d_in[2];   // [H,U]
    const float* att_neigh = (const float*)d_in[3];   // [H,U]
    const float* biases    = (const float*)d_in[4];   // [H,U]
    float* out = (float*)d_out;                       // [B,N,H*U]

    float* feats = (float*)d_ws;                      // [B*H][N/2][U][2] (8 MB)
    float* sbuf  = feats + (size_t)B_ * H_ * N_ * U_;
    float* tbuf  = sbuf + (size_t)B_ * H_ * N_;
    float* tmx   = tbuf + (size_t)B_ * H_ * N_;

    const dim3 gTile(N_ / 128, B_ * H_);
    gat_proj  <<<gTile, 256, 0, stream>>>(x, kernels, feats);
    gat_logits<<<(B_ * H_ * N_) / 256, 256, 0, stream>>>(feats, att_self, att_neigh, sbuf, tbuf);
    gat_tmax  <<<B_ * H_, 256, 0, stream>>>(tbuf, tmx);
    gat_attn  <<<gTile, 256, 0, stream>>>(feats, sbuf, tbuf, tmx, biases, out);
}